// MyEdgeWeightNorm_15977278341800
// MI455X (gfx1250) — compile-verified
//
#include <hip/hip_runtime.h>
#include <math.h>

// Edge-weight normalization (GCN norm='both') for MI455X (gfx1250, wave32).
//
//   out_deg[s] = sum_{e: src[e]==s} w[e]      (scatter-add, L2-resident table)
//   in_deg[d]  = sum_{e: dst[e]==d} w[e]
//   out[e]     = rsqrt(out_deg[src[e]]) * rsqrt(in_deg[dst[e]]) * w[e]
//
// Memory-bound: ~180 MB HBM traffic -> ~8 us at 23.3 TB/s.
//  * NT streaming loads for one-touch edge arrays (don't evict L2 table)
//  * explicit global_atomic_add_f32 (hardware f32 atomic at L2, no CAS loop)
//  * async global->LDS b128 staging of the weight stream in the gather kernel
//    (ASYNCcnt path), overlapping the copy with the norm-gather latency chain

typedef __attribute__((ext_vector_type(4))) float v4f;
typedef __attribute__((ext_vector_type(4))) int   v4i;

#define TB 256   // 8 wave32 waves per block

// Hardware f32 scatter-add, device scope, no return (STOREcnt).
__device__ __forceinline__ void atomic_add_f32_dev(float* __restrict__ base,
                                                   int idx, float val) {
    unsigned byte_off = (unsigned)idx << 2;
    asm volatile("global_atomic_add_f32 %0, %1, %2 scope:SCOPE_DEV"
                 :: "v"(byte_off), "v"(val), "s"(base)
                 : "memory");
}

// ---------------------------------------------------------------- zero degrees
__global__ void kzero_deg(float* __restrict__ deg, const int* __restrict__ n_ptr) {
    const int n2 = 2 * *n_ptr;   // scalar load, broadcast
    for (int i = blockIdx.x * blockDim.x + threadIdx.x; i < n2;
         i += gridDim.x * blockDim.x) {
        deg[i] = 0.0f;
    }
}

// ------------------------------------------------ scatter-add weighted degrees
__global__ void kaccum_deg(const float* __restrict__ w,
                           const int* __restrict__ src,
                           const int* __restrict__ dst,
                           float* __restrict__ deg,
                           const int* __restrict__ n_ptr,
                           int n_edges) {
    const int n = *n_ptr;
    float* __restrict__ out_deg = deg;       // indexed by src
    float* __restrict__ in_deg  = deg + n;   // indexed by dst

    const int q    = blockIdx.x * blockDim.x + threadIdx.x;
    const int base = q * 4;
    if (base + 3 < n_edges) {
        // NT streaming loads: one-touch data, keep L2 for the degree table
        v4f wv = __builtin_nontemporal_load((const v4f*)w   + q);
        v4i sv = __builtin_nontemporal_load((const v4i*)src + q);
        v4i dv = __builtin_nontemporal_load((const v4i*)dst + q);
#pragma unroll
        for (int k = 0; k < 4; ++k) {
            atomic_add_f32_dev(out_deg, sv[k], wv[k]);
            atomic_add_f32_dev(in_deg,  dv[k], wv[k]);
        }
    } else if (base < n_edges) {
        for (int e = base; e < n_edges; ++e) {
            const float we = w[e];
            atomic_add_f32_dev(out_deg, src[e], we);
            atomic_add_f32_dev(in_deg,  dst[e], we);
        }
    }
}

// ---------------------------------------------------------- deg -> rsqrt(deg)
__global__ void krsqrt_deg(float* __restrict__ deg, const int* __restrict__ n_ptr) {
    const int n2 = 2 * *n_ptr;
    for (int i = blockIdx.x * blockDim.x + threadIdx.x; i < n2;
         i += gridDim.x * blockDim.x) {
        deg[i] = rsqrtf(deg[i]);   // v_rsq_f32
    }
}

// --------------------------- out[e] = src_norm[src[e]] * dst_norm[dst[e]] * w[e]
// Weight quad is staged into LDS with an async b128 copy (ASYNCcnt) that
// overlaps the dependent index-load -> norm-gather chain.
__global__ void kedge_norm(const float* __restrict__ w,
                           const int* __restrict__ src,
                           const int* __restrict__ dst,
                           const float* __restrict__ deg,
                           float* __restrict__ out,
                           const int* __restrict__ n_ptr,
                           int n_edges) {
    __shared__ __align__(16) float wbuf[TB * 4];   // 4 KB of the 320 KB LDS

    const int n = *n_ptr;
    const float* __restrict__ src_norm = deg;
    const float* __restrict__ dst_norm = deg + n;

    const int t    = threadIdx.x;
    const int q    = blockIdx.x * blockDim.x + t;
    const int base = q * 4;

    if (base + 3 < n_edges) {
        // Kick off async copy: 16 B of w per lane, global -> LDS, no VGPR transit.
        const unsigned lds_off = (unsigned)(uintptr_t)(&wbuf[t * 4]);
        const unsigned g_off   = (unsigned)q * 16u;   // byte offset of this quad
        asm volatile("global_load_async_to_lds_b128 %0, %1, %2"
                     :: "v"(lds_off), "v"(g_off), "s"(w)
                     : "memory");

        // While the copy is in flight: index loads + L2-resident norm gathers.
        v4i sv = __builtin_nontemporal_load((const v4i*)src + q);
        v4i dv = __builtin_nontemporal_load((const v4i*)dst + q);
        float sn[4], dn[4];
#pragma unroll
        for (int k = 0; k < 4; ++k) {
            sn[k] = src_norm[sv[k]];
            dn[k] = dst_norm[dv[k]];
        }

        // Drain the async copy, then read staged weights back (ds_load_b128).
        asm volatile("s_wait_asynccnt 0x0" ::: "memory");
        v4f wv = *(const v4f*)(&wbuf[t * 4]);

        v4f r;
#pragma unroll
        for (int k = 0; k < 4; ++k) {
            r[k] = sn[k] * dn[k] * wv[k];
        }
        __builtin_nontemporal_store(r, (v4f*)out + q);   // one-touch output
    } else if (base < n_edges) {
        for (int e = base; e < n_edges; ++e) {
            out[e] = src_norm[src[e]] * dst_norm[dst[e]] * w[e];
        }
    }
}

// ------------------------------------------------------------------- launcher
extern "C" void kernel_launch(void* const* d_in, const int* in_sizes, int n_in,
                              void* d_out, int out_size, void* d_ws, size_t ws_size,
                              hipStream_t stream) {
    const float* w     = (const float*)d_in[0];
    const int*   src   = (const int*)  d_in[1];
    const int*   dst   = (const int*)  d_in[2];
    const int*   n_ptr = (const int*)  d_in[3];   // device scalar (num_nodes)

    float* deg = (float*)d_ws;    // [2 * num_nodes] floats (800 KB for N=100K)
    float* out = (float*)d_out;

    const int n_edges = in_sizes[0];
    const int quads   = (n_edges + 3) / 4;
    const int gq      = (quads + TB - 1) / TB;

    // num_nodes is only known on-device -> fixed-grid stride loops for node kernels
    kzero_deg <<<1024, TB, 0, stream>>>(deg, n_ptr);
    kaccum_deg<<<gq,   TB, 0, stream>>>(w, src, dst, deg, n_ptr, n_edges);
    krsqrt_deg<<<1024, TB, 0, stream>>>(deg, n_ptr);
    kedge_norm<<<gq,   TB, 0, stream>>>(w, src, dst, deg, out, n_ptr, n_edges);
}